// GNNTower_79508434583618
// MI455X (gfx1250) — compile-verified
//
#include <hip/hip_runtime.h>
#include <hip/hip_bf16.h>

typedef __attribute__((ext_vector_type(2))) float v2f;
typedef __attribute__((ext_vector_type(8))) float v8f;

#define DIM 128

// ---------------------------------------------------------------------------
// deg[n] = number of incoming edges (as float)
// ---------------------------------------------------------------------------
__global__ __launch_bounds__(256) void compute_deg(const int* __restrict__ dst,
                                                   float* __restrict__ deg,
                                                   int nEdges) {
  int e = blockIdx.x * blockDim.x + threadIdx.x;
  if (e < nEdges) unsafeAtomicAdd(&deg[dst[e]], 1.0f);
}

// ---------------------------------------------------------------------------
// agg[dst] += x[src]  (one wave per edge, lane handles 4 consecutive floats)
// ---------------------------------------------------------------------------
__global__ __launch_bounds__(256) void scatter_edges(const float* __restrict__ x,
                                                     const int* __restrict__ src,
                                                     const int* __restrict__ dst,
                                                     float* __restrict__ agg,
                                                     int nEdges) {
  int e = (int)((blockIdx.x * blockDim.x + threadIdx.x) >> 5);
  int lane = threadIdx.x & 31;
  if (e >= nEdges) return;
  int s = src[e];
  int d = dst[e];
  const float4 v = *(const float4*)(x + (size_t)s * DIM + lane * 4);
  float* ad = agg + (size_t)d * DIM + lane * 4;
  unsafeAtomicAdd(ad + 0, v.x);
  unsafeAtomicAdd(ad + 1, v.y);
  unsafeAtomicAdd(ad + 2, v.z);
  unsafeAtomicAdd(ad + 3, v.w);
}

// ---------------------------------------------------------------------------
// Fused layer: out = relu(x @ Wself + bself + (agg/deg) @ Wneigh + bneigh)
// 8 waves per block; wave w owns output columns [16w, 16w+16).
// K=128 split into 32 steps of V_WMMA_F32_16X16X4_F32.
// A layout (16x4 f32): lanes 0-15 -> M=lane, VGPR0=K0, VGPR1=K1;
//                      lanes 16-31 -> M=lane-16, VGPR0=K2, VGPR1=K3.
// B layout (4x16 f32): mirrors A over K (VGPR0: K0/K2, VGPR1: K1/K3, N=lane&15).
// C/D layout (16x16 f32, 8 VGPRs): VGPR r -> M = r + 8*(lane>=16), N = lane&15.
// ---------------------------------------------------------------------------
__global__ __launch_bounds__(256) void gemm_fused(
    const float* __restrict__ x, const float* __restrict__ agg,
    const float* __restrict__ deg,
    const float* __restrict__ Wself, const float* __restrict__ bself,
    const float* __restrict__ Wneigh, const float* __restrict__ bneigh,
    float* __restrict__ out, int nRows, int numTiles) {
  const int wave = threadIdx.x >> 5;   // 0..7 : column tile
  const int lane = threadIdx.x & 31;
  const int lm = lane & 15;
  const int hi = lane >> 4;            // 0 or 1
  const int n0 = wave * 16;
  const int col = n0 + lm;             // output feature this lane feeds for B
  const int kbase = hi * 2;            // K offset within each group of 4

  // Preload B fragments (both weight matrices, all 32 K-steps) into registers.
  v2f bs[32], bn[32];
#pragma unroll
  for (int kk = 0; kk < 32; ++kk) {
    const int k = kk * 4 + kbase;
    bs[kk].x = Wself[(size_t)k * DIM + col];
    bs[kk].y = Wself[(size_t)(k + 1) * DIM + col];
    bn[kk].x = Wneigh[(size_t)k * DIM + col];
    bn[kk].y = Wneigh[(size_t)(k + 1) * DIM + col];
  }
  const float bias = bself[col] + bneigh[col];

  for (int tile = blockIdx.x; tile < numTiles; tile += gridDim.x) {
    const int m0 = tile * 16;
    const int row = min(m0 + lm, nRows - 1);   // clamp (keeps EXEC all-ones)
    const float invdeg = 1.0f / fmaxf(deg[row], 1.0f);
    const float* xr = x + (size_t)row * DIM + kbase;
    const float* ar = agg + (size_t)row * DIM + kbase;

    v8f acc = {};
#pragma unroll
    for (int kk = 0; kk < 32; ++kk) {
      v2f a = *(const v2f*)(xr + kk * 4);
      acc = __builtin_amdgcn_wmma_f32_16x16x4_f32(
          false, a, false, bs[kk], (short)0, acc, false, false);
      v2f g = *(const v2f*)(ar + kk * 4);
      g.x *= invdeg;
      g.y *= invdeg;
      acc = __builtin_amdgcn_wmma_f32_16x16x4_f32(
          false, g, false, bn[kk], (short)0, acc, false, false);
    }

#pragma unroll
    for (int r = 0; r < 8; ++r) {
      const int mglob = m0 + r + hi * 8;
      float v = acc[r] + bias;
      v = v > 0.0f ? v : 0.0f;
      if (mglob < nRows) out[(size_t)mglob * DIM + n0 + lm] = v;
    }
  }
}

// ---------------------------------------------------------------------------
// Graph pooling: pooled[batch[n]] += x[n]; cnt[batch[n]] += 1
// ---------------------------------------------------------------------------
__global__ __launch_bounds__(256) void pool_nodes(const float* __restrict__ x,
                                                  const int* __restrict__ batch,
                                                  float* __restrict__ pooled,
                                                  float* __restrict__ cnt,
                                                  int nNodes) {
  int n = (int)((blockIdx.x * blockDim.x + threadIdx.x) >> 5);
  int lane = threadIdx.x & 31;
  if (n >= nNodes) return;
  int b = batch[n];
  const float4 v = *(const float4*)(x + (size_t)n * DIM + lane * 4);
  float* pd = pooled + (size_t)b * DIM + lane * 4;
  unsafeAtomicAdd(pd + 0, v.x);
  unsafeAtomicAdd(pd + 1, v.y);
  unsafeAtomicAdd(pd + 2, v.z);
  unsafeAtomicAdd(pd + 3, v.w);
  if (lane == 0) unsafeAtomicAdd(&cnt[b], 1.0f);
}

__global__ __launch_bounds__(256) void finalize_pool(const float* __restrict__ pooled,
                                                     const float* __restrict__ cnt,
                                                     float* __restrict__ out, int n) {
  int i = blockIdx.x * blockDim.x + threadIdx.x;
  if (i < n) out[i] = pooled[i] / fmaxf(cnt[i >> 7], 1.0f);
}

// ---------------------------------------------------------------------------
extern "C" void kernel_launch(void* const* d_in, const int* in_sizes, int n_in,
                              void* d_out, int out_size, void* d_ws, size_t ws_size,
                              hipStream_t stream) {
  const float* x_in    = (const float*)d_in[0];
  const int*   eidx    = (const int*)d_in[1];   // [2,E]: row 0 = src, row 1 = dst
  const int*   batch   = (const int*)d_in[2];
  const float* WsSelf  = (const float*)d_in[3];
  const float* bsSelf  = (const float*)d_in[4];
  const float* WsNeigh = (const float*)d_in[5];
  const float* bsNeigh = (const float*)d_in[6];
  float* out = (float*)d_out;

  const int N  = in_sizes[0] / DIM;            // 100000
  const int E  = in_sizes[1] / 2;              // 600000
  const int Bg = out_size / DIM;               // 64
  const int L  = in_sizes[3] / (DIM * DIM);    // 3
  const int numTiles = (N + 15) / 16;

  const int* src = eidx;
  const int* dst = eidx + E;

  // Workspace layout (all fits in ws; whole working set L2-resident).
  char* ws = (char*)d_ws;
  const size_t nd = (size_t)N * DIM * sizeof(float);
  float* bufA   = (float*)ws;
  float* bufB   = (float*)(ws + nd);
  float* agg    = (float*)(ws + 2 * nd);
  float* deg    = (float*)(ws + 3 * nd);
  size_t degBytes = (((size_t)N * sizeof(float)) + 255) & ~(size_t)255;
  float* pooled = (float*)(ws + 3 * nd + degBytes);
  float* cnt    = pooled + (size_t)Bg * DIM;

  // In-degree (constant across layers).
  hipMemsetAsync(deg, 0, (size_t)N * sizeof(float), stream);
  compute_deg<<<(E + 255) / 256, 256, 0, stream>>>(dst, deg, E);

  const float* cur = x_in;
  float* nxt = bufA;
  for (int l = 0; l < L; ++l) {
    hipMemsetAsync(agg, 0, nd, stream);
    long long sthreads = (long long)E * 32;
    scatter_edges<<<(int)((sthreads + 255) / 256), 256, 0, stream>>>(
        cur, src, dst, agg, E);
    gemm_fused<<<512, 256, 0, stream>>>(
        cur, agg, deg,
        WsSelf + (size_t)l * DIM * DIM, bsSelf + (size_t)l * DIM,
        WsNeigh + (size_t)l * DIM * DIM, bsNeigh + (size_t)l * DIM,
        nxt, N, numTiles);
    cur = nxt;
    nxt = (nxt == bufA) ? bufB : bufA;
  }

  hipMemsetAsync(pooled, 0, ((size_t)Bg * DIM + Bg) * sizeof(float), stream);
  long long pthreads = (long long)N * 32;
  pool_nodes<<<(int)((pthreads + 255) / 256), 256, 0, stream>>>(
      cur, batch, pooled, cnt, N);
  finalize_pool<<<(Bg * DIM + 255) / 256, 256, 0, stream>>>(pooled, cnt, out,
                                                            Bg * DIM);
}